// EdgeMidpointNodeScalar_89902255440814
// MI455X (gfx1250) — compile-verified
//
#include <hip/hip_runtime.h>

typedef float v2f __attribute__((ext_vector_type(2)));
typedef float v8f __attribute__((ext_vector_type(8)));

#define SCALAR 48
#define VEC    12
#define HID    96
#define LAYERS 3
#define E_IN   98     // 2*SCALAR + 2
#define N_IN   156    // SCALAR + HID + VEC
#define NB     2
#define NN     512
#define EPSF   1e-8f

// silu via v_exp_f32 + v_rcp_f32 (avoids the full IEEE divide sequence)
__device__ __forceinline__ float silu_f(float x) {
    float e = __expf(-x);
    return x * __builtin_amdgcn_rcpf(1.0f + e);
}

// ---------------------------------------------------------------------------
// Kernel 1: initialize h (broadcast h0) and zero v
// ---------------------------------------------------------------------------
__global__ void init_kernel(const float* __restrict__ h0,
                            float* __restrict__ h, float* __restrict__ v) {
    int tid = blockIdx.x * blockDim.x + threadIdx.x;
    if (tid < NB * NN * SCALAR) h[tid] = h0[tid % SCALAR];
    if (tid < NB * NN * VEC * 3) v[tid] = 0.0f;
}

// ---------------------------------------------------------------------------
// Kernel 2: per-node precompute  hA = h @ We1[l,0:48,:],  hB = h @ We1[l,48:96,:]
// grid = NB*NN blocks, 96 threads
// ---------------------------------------------------------------------------
__global__ void nodepre_kernel(const float* __restrict__ h,
                               const float* __restrict__ We1,
                               float* __restrict__ hA, float* __restrict__ hB, int l) {
    int node = blockIdx.x;          // b*NN + n
    int ch   = threadIdx.x;         // 0..95
    const float* W  = We1 + (size_t)l * E_IN * HID;
    const float* hn = h + (size_t)node * SCALAR;
    float a = 0.0f, b = 0.0f;
#pragma unroll
    for (int k = 0; k < SCALAR; ++k) {
        float hv = hn[k];
        a += hv * W[k * HID + ch];
        b += hv * W[(SCALAR + k) * HID + ch];
    }
    hA[(size_t)node * HID + ch] = a;
    hB[(size_t)node * HID + ch] = b;
}

// ---------------------------------------------------------------------------
// Kernel 3: edge kernel — one block per (b,i). 6 waves (192 threads).
// Loops j in tiles of 16. Per tile:
//   pre  = hA[i] + hB[j] + d2*wd + m2*wm + be1       (VALU, LDS sA <- silu(pre))
//   M    = silu(sA @ We2 + be2) * mask(j!=i)          (WMMA f32 16x16x4, K=96)
//   w    = M @ Wv                                     (VALU from LDS, small)
//   v[i] += sum_j w[j,:] (x) rel_dir[j,:]             (VALU accum in regs)
//   m_agg[i] += sum_j M[j,:]                          (VALU accum in regs)
// ---------------------------------------------------------------------------
__global__ void __launch_bounds__(192) edge_kernel(
    const float* __restrict__ pos,
    const float* __restrict__ hA, const float* __restrict__ hB,
    const float* __restrict__ We1, const float* __restrict__ be1,
    const float* __restrict__ We2, const float* __restrict__ be2,
    const float* __restrict__ Wv,
    float* __restrict__ v_out, float* __restrict__ m_agg, int l)
{
    __shared__ float sA[16 * HID];        // silu(pre) tile   (A matrix, 16 x 96)
    __shared__ float sM[16 * HID];        // masked silu(GEMM) (16 x 96)
    __shared__ float sW[16 * VEC];        // w = sM @ Wv
    __shared__ float sWv[HID * VEC];      // staged Wv[l]
    __shared__ float sd2[16], sm2[16], srel[16 * 3];

    const int node = blockIdx.x;          // b*NN + i
    const int b = node / NN, i = node % NN;
    const int tid  = threadIdx.x;
    const int lane = tid & 31;
    const int wave = tid >> 5;            // 0..5 -> output column block
    const int ln15 = lane & 15;
    const int khalf = (lane >> 4) * 2;    // 0 or 2

    const float* We1l = We1 + (size_t)l * E_IN * HID;
    const float* be1l = be1 + (size_t)l * HID;
    const float* We2l = We2 + (size_t)l * HID * HID;
    const float* be2l = be2 + (size_t)l * HID;
    const float* Wvl  = Wv  + (size_t)l * HID * VEC;

    // Stage Wv into LDS once (96x12 floats, 6 per thread)
#pragma unroll
    for (int s = 0; s < (HID * VEC) / 192; ++s)
        sWv[tid + s * 192] = Wvl[tid + s * 192];

    // Preload B-matrix fragments of We2 for this wave's 16-column block.
    // b[v] = We2[K = kc*4 + khalf + v, N = wave*16 + ln15]
    v2f bfrag[24];
#pragma unroll
    for (int kc = 0; kc < 24; ++kc) {
        int k0 = kc * 4 + khalf;
        bfrag[kc].x = We2l[(k0 + 0) * HID + wave * 16 + ln15];
        bfrag[kc].y = We2l[(k0 + 1) * HID + wave * 16 + ln15];
    }

    const float xi0 = pos[((size_t)b * NN + i) * 3 + 0];
    const float xi1 = pos[((size_t)b * NN + i) * 3 + 1];
    const float xi2 = pos[((size_t)b * NN + i) * 3 + 2];

    // per-thread constants for the pre phase: channel = tid % 96, row base = tid / 96
    const int ch   = tid % HID;
    const int jr0  = tid / HID;           // 0 or 1
    const float hAi  = hA[(size_t)node * HID + ch];
    const float wd   = We1l[(2 * SCALAR + 0) * HID + ch];   // d2 row
    const float wm   = We1l[(2 * SCALAR + 1) * HID + ch];   // m2 row
    const float be1v = be1l[ch];
    const float be2v = be2l[wave * 16 + ln15];

    float magg = 0.0f;                    // tid < 96: channel tid
    float vacc = 0.0f;                    // tid < 36: (c = tid/3, d = tid%3)
    const int vc = tid / 3, vd = tid % 3;

    for (int jt = 0; jt < NN / 16; ++jt) {
        const int jbase = jt * 16;

        // prefetch next tile's hB rows (overlaps with this tile's compute)
        if (jt + 1 < NN / 16)
            __builtin_prefetch(&hB[((size_t)b * NN + jbase + 16 + jr0) * HID + ch], 0, 1);

        // geometry for this j tile
        if (tid < 16) {
            int j = jbase + tid;
            float x0 = pos[((size_t)b * NN + j) * 3 + 0];
            float x1 = pos[((size_t)b * NN + j) * 3 + 1];
            float x2 = pos[((size_t)b * NN + j) * 3 + 2];
            float r0 = xi0 - x0, r1 = xi1 - x1, r2 = xi2 - x2;
            float d2 = r0 * r0 + r1 * r1 + r2 * r2;
            float md0 = 0.5f * (xi0 + x0), md1 = 0.5f * (xi1 + x1), md2 = 0.5f * (xi2 + x2);
            sd2[tid] = d2;
            sm2[tid] = md0 * md0 + md1 * md1 + md2 * md2;
            float inv = rsqrtf(d2 + EPSF);
            srel[tid * 3 + 0] = r0 * inv;
            srel[tid * 3 + 1] = r1 * inv;
            srel[tid * 3 + 2] = r2 * inv;
        }
        __syncthreads();

        // build sA = silu(first edge-MLP layer), 16x96, 8 rows per thread
#pragma unroll
        for (int s = 0; s < 8; ++s) {
            int j = jr0 + 2 * s;
            float pre = hAi + hB[((size_t)b * NN + jbase + j) * HID + ch]
                      + sd2[j] * wd + sm2[j] * wm + be1v;
            sA[j * HID + ch] = silu_f(pre);
        }
        __syncthreads();

        // 16x96 @ 96x16 GEMM via WMMA f32 16x16x4, K accumulated over 24 chunks
        v8f acc = {0.f, 0.f, 0.f, 0.f, 0.f, 0.f, 0.f, 0.f};
#pragma unroll
        for (int kc = 0; kc < 24; ++kc) {
            v2f a;
            a.x = sA[ln15 * HID + kc * 4 + khalf + 0];
            a.y = sA[ln15 * HID + kc * 4 + khalf + 1];
            acc = __builtin_amdgcn_wmma_f32_16x16x4_f32(
                false, a, false, bfrag[kc], (short)0, acc, false, false);
        }

        // write masked silu(acc + be2) to sM
        const int col = wave * 16 + ln15;
#pragma unroll
        for (int r = 0; r < 8; ++r) {
            int m = r + ((lane >> 4) << 3);       // row within tile
            int j = jbase + m;
            float val = (j == i) ? 0.0f : silu_f(acc[r] + be2v);
            sM[m * HID + col] = val;
        }
        __syncthreads();

        // w = sM @ Wv  (16 x 12), one (j,c) per thread, all operands in LDS
        {
            int j = tid / VEC, c = tid % VEC;     // 192 = 16*12 exactly
            float s = 0.0f;
#pragma unroll
            for (int k = 0; k < HID; ++k) s += sM[j * HID + k] * sWv[k * VEC + c];
            sW[j * VEC + c] = s;
        }
        __syncthreads();

        // accumulate m_agg and the equivariant vector update
        if (tid < HID) {
#pragma unroll
            for (int j = 0; j < 16; ++j) magg += sM[j * HID + tid];
        }
        if (tid < VEC * 3) {
#pragma unroll
            for (int j = 0; j < 16; ++j) vacc += sW[j * VEC + vc] * srel[j * 3 + vd];
        }
        __syncthreads();
    }

    if (tid < HID) m_agg[(size_t)node * HID + tid] = magg;
    if (tid < VEC * 3) v_out[(size_t)node * VEC * 3 + tid] += vacc;
}

// ---------------------------------------------------------------------------
// Kernel 4: node update. grid = NB*NN blocks, 96 threads.
// h += silu([h, m_agg, vnorm] @ Wn1 + bn1) @ Wn2 + bn2   (in place on h)
// ---------------------------------------------------------------------------
__global__ void nodeupd_kernel(float* __restrict__ h,
                               const float* __restrict__ m_agg,
                               const float* __restrict__ v,
                               const float* __restrict__ Wn1, const float* __restrict__ bn1,
                               const float* __restrict__ Wn2, const float* __restrict__ bn2,
                               int l)
{
    int node = blockIdx.x;
    int tid  = threadIdx.x;               // 0..95
    __shared__ float sn[N_IN];
    __shared__ float st[HID];

    if (tid < SCALAR) sn[tid] = h[(size_t)node * SCALAR + tid];
    sn[SCALAR + tid] = m_agg[(size_t)node * HID + tid];
    if (tid < VEC) {
        const float* vp = v + (size_t)node * VEC * 3 + tid * 3;
        sn[SCALAR + HID + tid] = sqrtf(vp[0] * vp[0] + vp[1] * vp[1] + vp[2] * vp[2] + EPSF);
    }
    __syncthreads();

    const float* W1 = Wn1 + (size_t)l * N_IN * HID;
    float acc = bn1[l * HID + tid];
#pragma unroll 4
    for (int k = 0; k < N_IN; ++k) acc += sn[k] * W1[k * HID + tid];
    st[tid] = silu_f(acc);
    __syncthreads();

    if (tid < SCALAR) {
        const float* W2 = Wn2 + (size_t)l * HID * SCALAR;
        float acc2 = bn2[l * SCALAR + tid];
#pragma unroll 4
        for (int k = 0; k < HID; ++k) acc2 += st[k] * W2[k * SCALAR + tid];
        h[(size_t)node * SCALAR + tid] += acc2;
    }
}

// ---------------------------------------------------------------------------
// Kernel 5: readout  out[b,n] = h @ Ws + bs
// ---------------------------------------------------------------------------
__global__ void readout_kernel(const float* __restrict__ h,
                               const float* __restrict__ Ws, const float* __restrict__ bs,
                               float* __restrict__ out) {
    int idx = blockIdx.x * blockDim.x + threadIdx.x;
    if (idx < NB * NN) {
        float acc = bs[0];
#pragma unroll
        for (int k = 0; k < SCALAR; ++k) acc += h[(size_t)idx * SCALAR + k] * Ws[k];
        out[idx] = acc;
    }
}

// ---------------------------------------------------------------------------
extern "C" void kernel_launch(void* const* d_in, const int* in_sizes, int n_in,
                              void* d_out, int out_size, void* d_ws, size_t ws_size,
                              hipStream_t stream) {
    const float* pos = (const float*)d_in[0];
    const float* h0  = (const float*)d_in[1];
    const float* We1 = (const float*)d_in[2];
    const float* be1 = (const float*)d_in[3];
    const float* We2 = (const float*)d_in[4];
    const float* be2 = (const float*)d_in[5];
    const float* Wv  = (const float*)d_in[6];
    const float* Wn1 = (const float*)d_in[7];
    const float* bn1 = (const float*)d_in[8];
    const float* Wn2 = (const float*)d_in[9];
    const float* bn2 = (const float*)d_in[10];
    const float* Ws  = (const float*)d_in[11];
    const float* bs  = (const float*)d_in[12];
    float* out = (float*)d_out;

    float* ws = (float*)d_ws;
    float* h     = ws;                              // NB*NN*SCALAR = 49152
    float* v     = h + NB * NN * SCALAR;            // NB*NN*VEC*3  = 36864
    float* hA    = v + NB * NN * VEC * 3;           // NB*NN*HID    = 98304
    float* hB    = hA + NB * NN * HID;              // NB*NN*HID
    float* magg  = hB + NB * NN * HID;              // NB*NN*HID

    init_kernel<<<(NB * NN * SCALAR + 255) / 256, 256, 0, stream>>>(h0, h, v);

    for (int l = 0; l < LAYERS; ++l) {
        nodepre_kernel<<<NB * NN, HID, 0, stream>>>(h, We1, hA, hB, l);
        edge_kernel<<<NB * NN, 192, 0, stream>>>(pos, hA, hB, We1, be1, We2, be2,
                                                 Wv, v, magg, l);
        nodeupd_kernel<<<NB * NN, HID, 0, stream>>>(h, magg, v, Wn1, bn1, Wn2, bn2, l);
    }

    readout_kernel<<<(NB * NN + 255) / 256, 256, 0, stream>>>(h, Ws, bs, out);
}